// RQVAE_22179211116771
// MI455X (gfx1250) — compile-verified
//
#include <hip/hip_runtime.h>

typedef float v8f __attribute__((ext_vector_type(8)));
typedef float v2f __attribute__((ext_vector_type(2)));

#define D_IN 512
#define D_H  128
#define KCB  256
#define BETA 0.25f

static __device__ __forceinline__ v8f wmma_f32(v2f a, v2f b, v8f c) {
  // (neg_a, A, neg_b, B, c_mod, C, reuse_a, reuse_b)
  return __builtin_amdgcn_wmma_f32_16x16x4_f32(false, a, false, b, (short)0, c,
                                               false, false);
}

// ---------------------------------------------------------------- init ------
__global__ void rq_init(float* __restrict__ acc, int* __restrict__ used) {
  int i = threadIdx.x;
  if (i < 4) acc[i] = 0.0f;
  if (i < 3 * KCB) used[i] = 0;
}

// ---------------------------------------------------------------- encode ----
// latent[N,128] = emb[N,512] @ W_enc[128,512]^T   (one wave = 16 rows)
__global__ __launch_bounds__(64) void rq_encode(const float* __restrict__ emb,
                                                const float* __restrict__ Wenc,
                                                float* __restrict__ L) {
  const int wave = threadIdx.x >> 5;
  const int lane = threadIdx.x & 31;
  const int l16  = lane & 15;
  const int half = lane >> 4;
  const long row0 = ((long)blockIdx.x * 2 + wave) * 16;

  const float* arow = emb + (row0 + l16) * D_IN + half * 2;
  v8f acc[8];
#pragma unroll
  for (int h = 0; h < 8; ++h) acc[h] = (v8f){};
  for (int kc = 0; kc < D_IN / 4; ++kc) {
    v2f a = *(const v2f*)(arow + kc * 4);
#pragma unroll
    for (int h = 0; h < 8; ++h) {
      v2f b = *(const v2f*)(Wenc + (long)(h * 16 + l16) * D_IN + half * 2 + kc * 4);
      acc[h] = wmma_f32(a, b, acc[h]);
    }
  }
#pragma unroll
  for (int h = 0; h < 8; ++h) {
    float* out = L + row0 * D_H + h * 16 + l16;
#pragma unroll
    for (int j = 0; j < 8; ++j)
      out[(long)(j + half * 8) * D_H] = acc[h][j];
  }
}

// ---------------------------------------------------------------- quantize --
// 3 residual VQ stages for a 16-row tile per wave; remainder lives in LDS.
__global__ __launch_bounds__(64) void rq_quant(float* __restrict__ L,
                                               const float* __restrict__ cb0,
                                               const float* __restrict__ cb1,
                                               const float* __restrict__ cb2,
                                               float* __restrict__ acc,
                                               int* __restrict__ used) {
  __shared__ float cnorm[3][KCB];
  __shared__ __align__(16) float dist[2][16][KCB];
  __shared__ __align__(16) float Rt[2][16][D_H];

  const int tid  = threadIdx.x;
  const int wave = tid >> 5, lane = tid & 31, l16 = lane & 15, half = lane >> 4;
  const float* cbs[3] = {cb0, cb1, cb2};

  // codebook squared norms (shared across both waves)
  for (int i = tid; i < 3 * KCB; i += 64) {
    const float* c = cbs[i >> 8] + (long)(i & 255) * D_H;
    float s = 0.0f;
    for (int d = 0; d < D_H; d += 4) {
      float4 v = *(const float4*)(c + d);
      s += v.x * v.x + v.y * v.y + v.z * v.z + v.w * v.w;
    }
    cnorm[i >> 8][i & 255] = s;
  }

  const long row0 = ((long)blockIdx.x * 2 + wave) * 16;
  // latent tile -> LDS (becomes the running remainder)
  for (int i = lane; i < 16 * (D_H / 4); i += 32) {
    int m = i >> 5, c4 = i & 31;
    ((float4*)&Rt[wave][m][0])[c4] = ((const float4*)(L + (row0 + m) * D_H))[c4];
  }
  __syncthreads();

  for (int s = 0; s < 3; ++s) {
    const float* cb = cbs[s];
    // distances: d2' = ||c||^2 - 2 * (r . c)   (argmin-equivalent)
    for (int ct = 0; ct < KCB / 16; ++ct) {
      v8f d = {};
      const float* brow = cb + (long)(ct * 16 + l16) * D_H + half * 2;
      const float* arow = &Rt[wave][l16][half * 2];
      for (int kc = 0; kc < D_H / 4; ++kc) {
        v2f a = *(const v2f*)(arow + kc * 4);   // LDS
        v2f b = *(const v2f*)(brow + kc * 4);   // global (L2 resident)
        d = wmma_f32(a, b, d);
      }
#pragma unroll
      for (int j = 0; j < 8; ++j)
        dist[wave][j + half * 8][ct * 16 + l16] =
            cnorm[s][ct * 16 + l16] - 2.0f * d[j];
    }
    __syncthreads();

    // per-row argmin over 256 codes (first-min tie-break like jnp.argmin)
    int best[16];
#pragma unroll 1
    for (int m = 0; m < 16; ++m) {
      float bd = 3.4e38f;
      int bi = 0;
#pragma unroll
      for (int i = 0; i < 8; ++i) {
        int k = lane * 8 + i;
        float dv = dist[wave][m][k];
        if (dv < bd) { bd = dv; bi = k; }
      }
#pragma unroll
      for (int off = 16; off >= 1; off >>= 1) {
        float od = __shfl_xor(bd, off, 32);
        int   oi = __shfl_xor(bi, off, 32);
        if (od < bd || (od == bd && oi < bi)) { bd = od; bi = oi; }
      }
      best[m] = bi;
    }

    // remainder update + (1+beta)*mse accumulation
    float sumsq = 0.0f;
#pragma unroll 1
    for (int m = 0; m < 16; ++m) {
      int idx = best[m];
      if (lane == 0) used[s * KCB + idx] = 1;
      float4 r = ((float4*)&Rt[wave][m][0])[lane];
      float4 v = ((const float4*)(cb + (long)idx * D_H))[lane];
      float4 df;
      df.x = r.x - v.x; df.y = r.y - v.y; df.z = r.z - v.z; df.w = r.w - v.w;
      sumsq += df.x * df.x + df.y * df.y + df.z * df.z + df.w * df.w;
      ((float4*)&Rt[wave][m][0])[lane] = df;
    }
#pragma unroll
    for (int off = 16; off >= 1; off >>= 1) sumsq += __shfl_xor(sumsq, off, 32);
    if (lane == 0) atomicAdd(&acc[s], sumsq);
    __syncthreads();
  }

  // restored = latent - remainder, written back into L in place
  for (int i = lane; i < 16 * (D_H / 4); i += 32) {
    int m = i >> 5, c4 = i & 31;
    float4 lv = ((const float4*)(L + (row0 + m) * D_H))[c4];
    float4 rv = ((float4*)&Rt[wave][m][0])[c4];
    float4 o;
    o.x = lv.x - rv.x; o.y = lv.y - rv.y; o.z = lv.z - rv.z; o.w = lv.w - rv.w;
    ((float4*)(L + (row0 + m) * D_H))[c4] = o;
  }
}

// ---------------------------------------------------------------- decode ----
// fused: (restored @ W_dec^T - emb)^2 accumulated, output never materialized
__global__ __launch_bounds__(64) void rq_decode(const float* __restrict__ REST,
                                                const float* __restrict__ Wdec,
                                                const float* __restrict__ emb,
                                                float* __restrict__ acc) {
  const int wave = threadIdx.x >> 5, lane = threadIdx.x & 31;
  const int l16 = lane & 15, half = lane >> 4;
  const long row0 = ((long)blockIdx.x * 2 + wave) * 16;
  const float* arow = REST + (row0 + l16) * D_H + half * 2;
  float sumsq = 0.0f;
  for (int dt = 0; dt < D_IN / 16; ++dt) {
    v8f d = {};
    const float* brow = Wdec + (long)(dt * 16 + l16) * D_H + half * 2;
    for (int kc = 0; kc < D_H / 4; ++kc) {
      v2f a = *(const v2f*)(arow + kc * 4);
      v2f b = *(const v2f*)(brow + kc * 4);
      d = wmma_f32(a, b, d);
    }
#pragma unroll
    for (int j = 0; j < 8; ++j) {
      float e = emb[(row0 + j + half * 8) * D_IN + dt * 16 + l16];
      float df = d[j] - e;
      sumsq += df * df;
    }
  }
#pragma unroll
  for (int off = 16; off >= 1; off >>= 1) sumsq += __shfl_xor(sumsq, off, 32);
  if (lane == 0) atomicAdd(&acc[3], sumsq);
}

// ---------------------------------------------------------------- finalize --
__global__ void rq_final(const float* __restrict__ acc, const int* __restrict__ used,
                         float* __restrict__ out, int N) {
  if (threadIdx.x == 0 && blockIdx.x == 0) {
    float rq = (acc[0] + acc[1] + acc[2]) *
               ((1.0f + BETA) / ((float)N * (float)D_H));
    float recon = acc[3] / ((float)N * (float)D_IN);
    out[0] = recon + rq;
    out[1] = recon;
    out[2] = rq;
    for (int s = 0; s < 3; ++s) {
      int c = 0;
      for (int k = 0; k < KCB; ++k) c += used[s * KCB + k] ? 1 : 0;
      out[3 + s] = (float)c;
    }
  }
}

// ---------------------------------------------------------------- launch ----
extern "C" void kernel_launch(void* const* d_in, const int* in_sizes, int n_in,
                              void* d_out, int out_size, void* d_ws, size_t ws_size,
                              hipStream_t stream) {
  const float* emb  = (const float*)d_in[0];
  const float* Wenc = (const float*)d_in[1];
  const float* Wdec = (const float*)d_in[2];
  const float* cb0  = (const float*)d_in[3];
  const float* cb1  = (const float*)d_in[4];
  const float* cb2  = (const float*)d_in[5];
  float* out = (float*)d_out;
  const int N = in_sizes[0] / D_IN;   // 131072

  float* L    = (float*)d_ws;                 // N x 128 latent / restored
  float* acc  = L + (size_t)N * D_H;          // 4 loss accumulators
  int*   used = (int*)(acc + 4);              // 3 x 256 used-code flags

  hipLaunchKernelGGL(rq_init, dim3(1), dim3(1024), 0, stream, acc, used);
  const int nblk = N / 32;                    // 2 waves x 16 rows per block
  hipLaunchKernelGGL(rq_encode, dim3(nblk), dim3(64), 0, stream, emb, Wenc, L);
  hipLaunchKernelGGL(rq_quant,  dim3(nblk), dim3(64), 0, stream, L, cb0, cb1, cb2, acc, used);
  hipLaunchKernelGGL(rq_decode, dim3(nblk), dim3(64), 0, stream, L, Wdec, emb, acc);
  hipLaunchKernelGGL(rq_final,  dim3(1), dim3(32), 0, stream, acc, used, out, N);
}